// NTF_4209067950363
// MI455X (gfx1250) — compile-verified
//
#include <hip/hip_runtime.h>
#include <hip/hip_bf16.h>

// CP reconstruction as GEMM: D[M=65536, N=1024] = A[M,32] x B[32,N]
//   A[(i*256+j), r] = W0[i,r] * W1[j,r]   (built in registers, per-wave)
//   B[r, k]         = W2[k, r]            (staged in LDS via TDM, 128 KB)
// Store-bandwidth-bound: 256 MB out @ 23.3 TB/s ~= 11.5 us floor.
// Full f32 precision via V_WMMA_F32_16X16X4_F32 (8 per 16x16 tile, K=32).

typedef __attribute__((ext_vector_type(2))) float        v2f;
typedef __attribute__((ext_vector_type(4))) float        v4f;
typedef __attribute__((ext_vector_type(8))) float        v8f;
typedef __attribute__((ext_vector_type(4))) unsigned int v4u;
typedef __attribute__((ext_vector_type(8))) int          v8i;
typedef __attribute__((ext_vector_type(4))) int          v4i;

#define DIM_I 256
#define DIM_J 256
#define DIM_K 1024
#define RANK  32

#if defined(__HIP_DEVICE_COMPILE__) && __has_builtin(__builtin_amdgcn_tensor_load_to_lds)
#define USE_TDM 1
#if __has_include(<hip/amd_detail/amd_gfx1250_TDM.h>)
#define TDM_6ARG 1   // therock-10.0 headers -> clang-23 6-arg builtin
#else
#define TDM_6ARG 0   // ROCm 7.2 clang-22 5-arg builtin
#endif
#else
#define USE_TDM 0
#endif

extern "C" __global__ void __launch_bounds__(256)
ntf_cp_wmma_f32(const float* __restrict__ W0,
                const float* __restrict__ W1,
                const float* __restrict__ W2,
                float* __restrict__ out)
{
    extern __shared__ float ldsW2[];   // DIM_K * RANK floats = 128 KB

    const int tid  = threadIdx.x;
    const int lane = tid & 31;         // wave32
    const int wave = tid >> 5;         // 8 waves per block
    const int half = lane >> 4;        // 0: lanes 0-15, 1: lanes 16-31
    const int l16  = lane & 15;

    // ---- Stage all of W2 (1024x32 f32 = 128 KB) into LDS ----
#if USE_TDM
    if (wave == 0) {
        // Tensor DMA descriptor (D#), CDNA5 ISA:
        // 1D tile: 32768 elements of 4 bytes, contiguous global -> LDS.
        const unsigned long long ga = (unsigned long long)(const void*)W2;
        const unsigned ga_lo = (unsigned)ga;
        const unsigned ga_hi = (unsigned)(ga >> 32);
        // Flat shared address: low 32 bits are the LDS byte offset.
        const unsigned lds_addr = (unsigned)(unsigned long long)(const void*)ldsW2;

        // Group 0: count=1 | lds_addr | global_addr[56:0] | type=2
        v4u g0 = { 1u,
                   lds_addr,
                   ga_lo,
                   (ga_hi & 0x01FFFFFFu) | (2u << 30) };
        // Group 1: data_size=2 (4B); tensor_dim0=32768; tensor_dim1=1;
        //          tile_dim0=32768; tile_dim1/2=0; tensor_dim0_stride=32768.
        v8i g1 = { (int)0x00020000u,   // [17:16] data_size=2 (4 bytes)
                   (int)0x80000000u,   // [63:48] tensor_dim0.lo16 = 32768
                   (int)0x00010000u,   // [95:80] tensor_dim1 = 1
                   (int)0x80000000u,   // [127:112] tile_dim0 = 32768
                   0,                  // tile_dim1 = 0 (unused), tile_dim2 = 0
                   32768,              // [191:160] tensor_dim0_stride.lo32
                   0, 0 };
        v4i g2 = { 0, 0, 0, 0 };
        v4i g3 = { 0, 0, 0, 0 };
#if TDM_6ARG
        v8i g4 = { 0, 0, 0, 0, 0, 0, 0, 0 };
        __builtin_amdgcn_tensor_load_to_lds(g0, g1, g2, g3, g4, 0);
#else
        __builtin_amdgcn_tensor_load_to_lds(g0, g1, g2, g3, 0);
#endif
#if __has_builtin(__builtin_amdgcn_s_wait_tensorcnt)
        __builtin_amdgcn_s_wait_tensorcnt(0);
#endif
    }
    __syncthreads();
#else
    for (int idx = tid * 4; idx < DIM_K * RANK; idx += 256 * 4) {
        *(v4f*)(ldsW2 + idx) = *(const v4f*)(W2 + idx);
    }
    __syncthreads();
#endif

    // ---- This wave's M-tile: 16 rows of A = (i fixed, j0..j0+15) ----
    const int mt = blockIdx.x * 8 + wave;   // 0..4095 M-tiles
    const int i  = mt >> 4;                 // 16 j-tiles per i
    const int j0 = (mt & 15) << 4;

    // ---- Build A fragments in registers (f32 16x4 layout, 8 K-chunks) ----
    // A frag chunk c: lanes 0-15 hold {K=4c, 4c+1} for row M=lane;
    //                 lanes 16-31 hold {K=4c+2, 4c+3} for row M=lane-16.
    const float* __restrict__ w0p = W0 + i * RANK;          // row of W0
    const float* __restrict__ w1p = W1 + (j0 + l16) * RANK; // per-lane row of W1
    v2f afrag[8];
#pragma unroll
    for (int c = 0; c < 8; ++c) {
        const int rb = 4 * c + 2 * half;
        v2f w0v = *(const v2f*)(w0p + rb);
        v2f w1v = *(const v2f*)(w1p + rb);
        afrag[c] = w0v * w1v;   // Khatri-Rao row element-wise product
    }

    float* __restrict__ outbase =
        out + (size_t)(i * DIM_J + j0) * DIM_K + (size_t)half * 8 * DIM_K + l16;

    // ---- Sweep N: 64 tiles of 16 columns each ----
    for (int nt = 0; nt < DIM_K / 16; ++nt) {
        const int k0 = nt * 16;

        // B frag chunk c (B = 4x16, rows=K): VGPR0 lanes 0-15 row K=4c,
        // lanes 16-31 row K=4c+2; VGPR1 rows 4c+1 / 4c+3.
        // B[r, k] = W2[k, r]  ->  per-lane v2f load from LDS.
        const float* __restrict__ bp = ldsW2 + (k0 + l16) * RANK + 2 * half;
        v2f bfrag[8];
#pragma unroll
        for (int c = 0; c < 8; ++c) {
            bfrag[c] = *(const v2f*)(bp + 4 * c);   // ds_load_2addr_b64
        }

        v8f acc = {};
#pragma unroll
        for (int c = 0; c < 8; ++c) {
            // 8 args: (neg_a, A, neg_b, B, c_mod, C, reuse_a, reuse_b)
            acc = __builtin_amdgcn_wmma_f32_16x16x4_f32(
                false, afrag[c], false, bfrag[c], (short)0, acc, false, false);
        }

        // D layout: VGPR v -> row M = v + 8*half, col N = l16.
        // Output is write-once / never re-read: non-temporal stores.
        float* __restrict__ op = outbase + k0;
#pragma unroll
        for (int v = 0; v < 8; ++v) {
            __builtin_nontemporal_store(acc[v], op + (size_t)v * DIM_K);
        }
    }
}

extern "C" void kernel_launch(void* const* d_in, const int* in_sizes, int n_in,
                              void* d_out, int out_size, void* d_ws, size_t ws_size,
                              hipStream_t stream) {
    const float* W0 = (const float*)d_in[0];   // (256, 32)
    const float* W1 = (const float*)d_in[1];   // (256, 32)
    const float* W2 = (const float*)d_in[2];   // (1024, 32)
    float* out = (float*)d_out;                // (256, 256, 1024)

    // 4096 M-tiles, 8 waves (one M-tile each) per 256-thread block.
    const int blocks = (DIM_I * DIM_J / 16) / 8;   // 512
    const size_t lds_bytes = (size_t)DIM_K * RANK * sizeof(float);  // 128 KB
    ntf_cp_wmma_f32<<<blocks, 256, lds_bytes, stream>>>(W0, W1, W2, out);
}